// HybridModel_13537736917412
// MI455X (gfx1250) — compile-verified
//
#include <hip/hip_runtime.h>
#include <math.h>

typedef float v2f __attribute__((ext_vector_type(2)));
typedef float v8f __attribute__((ext_vector_type(8)));

// ---------------------------------------------------------------------------
// Complex helpers (gate-prep only; tiny workload)
// ---------------------------------------------------------------------------
struct C2 { float re, im; };
__device__ __forceinline__ C2 cmul(C2 a, C2 b) {
    return { a.re * b.re - a.im * b.im, a.re * b.im + a.im * b.re };
}
__device__ __forceinline__ C2 cadd(C2 a, C2 b) { return { a.re + b.re, a.im + b.im }; }

__device__ void su2m(float a, float b, float c, C2 M[2][2]) {
    // rz(a) @ ry(b) @ rz(c) closed form
    float sb, cb; sincosf(0.5f * b, &sb, &cb);
    float sp, cp; sincosf(0.5f * (a + c), &sp, &cp);
    float sm, cm; sincosf(0.5f * (a - c), &sm, &cm);
    M[0][0] = {  cb * cp, -cb * sp };   //  cos(b/2) e^{-i(a+c)/2}
    M[0][1] = { -sb * cm,  sb * sm };   // -sin(b/2) e^{-i(a-c)/2}
    M[1][0] = {  sb * cm,  sb * sm };   //  sin(b/2) e^{+i(a-c)/2}
    M[1][1] = {  cb * cp,  cb * sp };   //  cos(b/2) e^{+i(a+c)/2}
}

__device__ void kron2(const C2 A[2][2], const C2 B[2][2], C2 K[4][4]) {
    for (int i = 0; i < 2; i++)
        for (int j = 0; j < 2; j++)
            for (int k = 0; k < 2; k++)
                for (int l = 0; l < 2; l++)
                    K[i * 2 + k][j * 2 + l] = cmul(A[i][j], B[k][l]);
}

__device__ void cmm4(const C2 A[4][4], const C2 B[4][4], C2 O[4][4]) {
    for (int i = 0; i < 4; i++)
        for (int j = 0; j < 4; j++) {
            C2 acc = { 0.f, 0.f };
            for (int k = 0; k < 4; k++) acc = cadd(acc, cmul(A[i][k], B[k][j]));
            O[i][j] = acc;
        }
}

__device__ void build_su4(const float* __restrict__ t, C2 G[4][4]) {
    C2 s1[2][2], s2[2][2], Bm[4][4], Am[4][4];
    su2m(t[0], t[1], t[2], s1);  su2m(t[3],  t[4],  t[5],  s2);  kron2(s1, s2, Bm);
    su2m(t[9], t[10], t[11], s1); su2m(t[12], t[13], t[14], s2); kron2(s1, s2, Am);

    float sx, cx, sy, cy, sz, cz;
    sincosf(t[6], &sx, &cx); sincosf(t[7], &sy, &cy); sincosf(t[8], &sz, &cz);
    C2 ex[4][4], ey[4][4], ez[4][4];
    for (int p = 0; p < 4; p++)
        for (int q = 0; q < 4; q++) { ex[p][q] = {0,0}; ey[p][q] = {0,0}; ez[p][q] = {0,0}; }
    for (int p = 0; p < 4; p++) {
        ex[p][p].re = cx;
        ey[p][p].re = cy;
        ex[p][3 - p].im = sx;                                   // i sin(x) * XX
        ey[p][3 - p].im = (p == 1 || p == 2) ? sy : -sy;        // i sin(y) * YY
        ez[p][p] = { cz, (p == 0 || p == 3) ? sz : -sz };       // exp(i z ZZ)
    }
    C2 m1[4][4], core[4][4], t1[4][4];
    cmm4(ex, ey, m1);
    cmm4(m1, ez, core);
    cmm4(core, Bm, t1);
    cmm4(Am, t1, G);
}

// ---------------------------------------------------------------------------
// Kernel 1: build 10 fused quad-gate matrices (real 32x32 each) into d_ws.
//   Layer l: s=0: U(l,0)⊗U(l,2)  bits 9..6
//            s=1: U(l,4)⊗U(l,6)  bits 5..2
//            s=2: I4   ⊗U(l,8)   bits 3..0
//            s=3: U(l,1)⊗U(l,3)  bits 8..5
//            s=4: U(l,5)⊗U(l,7)  bits 4..1
//   Real embedding rows/cols: [0..15]=Re, [16..31]=Im -> [[Re,-Im],[Im,Re]].
// ---------------------------------------------------------------------------
__global__ __launch_bounds__(128) void prep_gates(const float* __restrict__ su4w,
                                                  float* __restrict__ gates) {
    __shared__ C2 UG[18][16];
    const int tid = threadIdx.x;
    if (tid < 18) {
        C2 G[4][4];
        build_su4(su4w + tid * 15, G);
        for (int p = 0; p < 4; p++)
            for (int q = 0; q < 4; q++)
                UG[tid][p * 4 + q] = G[p][q];
    }
    __syncthreads();

    for (int e = tid; e < 10 * 1024; e += 128) {
        const int g = e >> 10;
        const int r = (e >> 5) & 31;
        const int c = e & 31;
        const int l = g / 5, s = g % 5;
        int gA, gB;
        switch (s) {
            case 0: gA = l * 9 + 0; gB = l * 9 + 2; break;
            case 1: gA = l * 9 + 4; gB = l * 9 + 6; break;
            case 2: gA = -1;        gB = l * 9 + 8; break;
            case 3: gA = l * 9 + 1; gB = l * 9 + 3; break;
            default:gA = l * 9 + 5; gB = l * 9 + 7; break;
        }
        const int p = r & 15, q = c & 15;
        const int pA = p >> 2, pB = p & 3, qA = q >> 2, qB = q & 3;
        C2 uA = (gA < 0) ? C2{ (pA == qA) ? 1.f : 0.f, 0.f } : UG[gA][pA * 4 + qA];
        C2 uB = UG[gB][pB * 4 + qB];
        C2 u  = cmul(uA, uB);
        float val;
        if (r < 16) val = (c < 16) ?  u.re : -u.im;   // [Re, -Im]
        else        val = (c < 16) ?  u.im :  u.re;   // [Im,  Re]
        gates[g * 1024 + r * 32 + c] = val;
    }
}

// ---------------------------------------------------------------------------
// Kernel 2: one wave per batch element; state (1024 cplx) lives in LDS.
// Each quad-gate: Y(32x64) = M32(32x32) @ S(32x64) via v_wmma_f32_16x16x4_f32.
// ---------------------------------------------------------------------------
#define WPB 4   // waves per block

__global__ __launch_bounds__(WPB * 32) void pqc_main(
    const float* __restrict__ x, const float* __restrict__ W1,
    const float* __restrict__ b1, const float* __restrict__ W2,
    const float* __restrict__ b2, const float* __restrict__ gates,
    float* __restrict__ out, int B) {

    __shared__ float sRe[WPB][1024];
    __shared__ float sIm[WPB][1024];
    __shared__ float sAng[WPB][10];

    const int lane  = threadIdx.x & 31;
    const int wv    = threadIdx.x >> 5;
    const int batch = blockIdx.x * WPB + wv;
    if (batch >= B) return;

    // ---- angles = pi * tanh(x @ W1 + b1) : 10 dots of length 64 ----
    if (lane < 10) {
        float acc = b1[lane];
        const float* xr = x + batch * 64;
        #pragma unroll 8
        for (int k = 0; k < 64; k++) acc += xr[k] * W1[k * 10 + lane];
        sAng[wv][lane] = 3.14159265358979323846f * tanhf(acc);
    }
    __syncthreads();

    // ---- initial product state (real), qubit q -> index bit (9-q) ----
    {
        float cs[10], sn[10];
        #pragma unroll
        for (int q = 0; q < 10; q++) {
            float s, c; __sincosf(0.5f * sAng[wv][q], &s, &c);
            cs[9 - q] = c; sn[9 - q] = s;
        }
        for (int m = 0; m < 32; m++) {
            int idx = m * 32 + lane;
            float p = 1.f;
            #pragma unroll
            for (int b = 0; b < 10; b++) p *= ((idx >> b) & 1) ? sn[b] : cs[b];
            sRe[wv][idx] = p;
            sIm[wv][idx] = 0.f;
        }
    }

    // ---- 10 quad-gates: 2 layers x (even s=0,1,2 ; odd s=3,4) ----
    const int tshift[5] = { 6, 2, 0, 5, 1 };
    const int am = lane & 15;        // A: M row
    const int ak = (lane >> 4) * 2;  // A/B: K sub-offset per half-wave
    const int cl = lane & 15;        // B/C: N column within tile

    for (int g = 0; g < 10; g++) {
        const int t = tshift[g % 5];
        const int lowmask = (1 << t) - 1;
        const float* A = gates + g * 1024;

        // A fragments: 2 row-tiles x 8 K-chunks (16x4 each)
        v2f a0[8], a1[8];
        #pragma unroll
        for (int k = 0; k < 8; k++) {
            a0[k] = *(const v2f*)(A + am * 32        + 4 * k + ak);
            a1[k] = *(const v2f*)(A + (16 + am) * 32 + 4 * k + ak);
        }

        #pragma unroll
        for (int ct = 0; ct < 4; ct++) {
            const int c    = ct * 16 + cl;
            const int base = ((c >> t) << (t + 4)) | (c & lowmask);

            // B fragments: gather S rows (Re rows 0..15, Im rows 16..31)
            v2f bf[8];
            #pragma unroll
            for (int k = 0; k < 8; k++) {
                int r0 = 4 * k + ak;
                const float* S = (k < 4) ? sRe[wv] : sIm[wv];
                int q0 = r0 & 15, q1 = (r0 + 1) & 15;
                v2f bb;
                bb.x = S[base | (q0 << t)];
                bb.y = S[base | (q1 << t)];
                bf[k] = bb;
            }

            v8f c0 = { 0, 0, 0, 0, 0, 0, 0, 0 };
            v8f c1 = { 0, 0, 0, 0, 0, 0, 0, 0 };
            #pragma unroll
            for (int k = 0; k < 8; k++) {
                c0 = __builtin_amdgcn_wmma_f32_16x16x4_f32(
                        false, a0[k], false, bf[k], (short)0, c0, false, false);
                c1 = __builtin_amdgcn_wmma_f32_16x16x4_f32(
                        false, a1[k], false, bf[k], (short)0, c1, false, false);
            }

            // scatter back in place (disjoint per column tile)
            const int rofs = (lane >> 4) * 8;
            #pragma unroll
            for (int r = 0; r < 8; r++) {
                int q = r + rofs;
                sRe[wv][base | (q << t)] = c0[r];   // rows 0..15  -> Re
                sIm[wv][base | (q << t)] = c1[r];   // rows 16..31 -> Im
            }
        }
    }

    // ---- <Z_q>: z[q] = sum p * (1 - 2*bit_{9-q}) ; then out = z @ W2 + b2 ----
    float z[10];
    #pragma unroll
    for (int b = 0; b < 10; b++) z[b] = 0.f;
    for (int m = 0; m < 32; m++) {
        int idx = m * 32 + lane;
        float re = sRe[wv][idx], im = sIm[wv][idx];
        float p = re * re + im * im;
        #pragma unroll
        for (int b = 0; b < 10; b++) z[b] += ((idx >> b) & 1) ? -p : p;
    }
    #pragma unroll
    for (int off = 16; off >= 1; off >>= 1) {
        #pragma unroll
        for (int b = 0; b < 10; b++) z[b] += __shfl_xor(z[b], off, 32);
    }
    if (lane == 0) {
        float r = b2[0];
        #pragma unroll
        for (int q = 0; q < 10; q++) r += z[9 - q] * W2[q];
        out[batch] = r;
    }
}

// ---------------------------------------------------------------------------
extern "C" void kernel_launch(void* const* d_in, const int* in_sizes, int n_in,
                              void* d_out, int out_size, void* d_ws, size_t ws_size,
                              hipStream_t stream) {
    const float* x    = (const float*)d_in[0];  // (B, 64)
    const float* W1   = (const float*)d_in[1];  // (64, 10)
    const float* b1   = (const float*)d_in[2];  // (10)
    const float* su4w = (const float*)d_in[3];  // (2, 9, 15)
    const float* W2   = (const float*)d_in[4];  // (10, 1)
    const float* b2   = (const float*)d_in[5];  // (1)
    float*       out  = (float*)d_out;          // (B, 1)
    float*       gws  = (float*)d_ws;           // 10 * 32 * 32 f32 = 40 KB

    const int B = in_sizes[0] / 64;

    prep_gates<<<1, 128, 0, stream>>>(su4w, gws);
    pqc_main<<<(B + WPB - 1) / WPB, WPB * 32, 0, stream>>>(x, W1, b1, W2, b2, gws, out, B);
}